// Net_67113158967656
// MI455X (gfx1250) — compile-verified
//
#include <hip/hip_runtime.h>

static constexpr int NN = 50000;   // nodes (3125 * 16, exact)
static constexpr int EE = 800000;  // edges
static constexpr int HH = 128;     // output channels of every GEMM

typedef __attribute__((ext_vector_type(2))) float v2f;
typedef __attribute__((ext_vector_type(8))) float v8f;

// ---------------- elementwise / scatter kernels ----------------

__global__ void fill_kernel(float* __restrict__ p, int n, float v) {
  int i = blockIdx.x * blockDim.x + threadIdx.x;
  if (i < n) p[i] = v;
}

__global__ void deg_scatter_kernel(const int* __restrict__ dst, float* __restrict__ deg) {
  int e = blockIdx.x * blockDim.x + threadIdx.x;
  if (e < EE) atomicAdd(&deg[dst[e]], 1.0f);
}

// agg[dst] += x[src], float4-vectorized over features
__global__ void sage_scatter_kernel(const int* __restrict__ src, const int* __restrict__ dst,
                                    const float* __restrict__ x, float* __restrict__ agg,
                                    int F4) {
  int i = blockIdx.x * blockDim.x + threadIdx.x;
  int total = EE * F4;
  if (i >= total) return;
  int e = i / F4;
  int j = (i - e * F4) * 4;
  int F = F4 * 4;
  int s = src[e], d = dst[e];
  const float4 v = *(const float4*)(x + (size_t)s * F + j);
  float* o = agg + (size_t)d * F + j;
  atomicAdd(o + 0, v.x);
  atomicAdd(o + 1, v.y);
  atomicAdd(o + 2, v.z);
  atomicAdd(o + 3, v.w);
}

__global__ void sage_div_kernel(float* __restrict__ agg, const float* __restrict__ deg, int F) {
  int i = blockIdx.x * blockDim.x + threadIdx.x;
  if (i >= NN * F) return;
  agg[i] /= fmaxf(deg[i / F], 1.0f);
}

// out[dst] += rsqrt(deg[src]*deg[dst]) * xw[src]  (xw is [N,128])
__global__ void gcn_scatter_kernel(const int* __restrict__ src, const int* __restrict__ dst,
                                   const float* __restrict__ xw, const float* __restrict__ deg,
                                   float* __restrict__ out) {
  int i = blockIdx.x * blockDim.x + threadIdx.x;
  int total = EE * 32;  // 128/4 groups per edge
  if (i >= total) return;
  int e = i >> 5;
  int j = (i & 31) * 4;
  int s = src[e], d = dst[e];
  float norm = rsqrtf(deg[s] * deg[d]);
  const float4 v = *(const float4*)(xw + (size_t)s * 128 + j);
  float* o = out + (size_t)d * 128 + j;
  atomicAdd(o + 0, norm * v.x);
  atomicAdd(o + 1, norm * v.y);
  atomicAdd(o + 2, norm * v.z);
  atomicAdd(o + 3, norm * v.w);
}

// out[:, co:co+128] = gtmp + xw/deg + b + res1 (+ res2)
__global__ void gcn_finish_kernel(const float* __restrict__ gtmp, const float* __restrict__ xw,
                                  const float* __restrict__ deg, const float* __restrict__ b,
                                  float* __restrict__ out, int ldc, int co,
                                  const float* __restrict__ res1, int ldr1,
                                  const float* __restrict__ res2, int ldr2) {
  int i = blockIdx.x * blockDim.x + threadIdx.x;
  if (i >= NN * 128) return;
  int node = i >> 7;
  int j = i & 127;
  float v = gtmp[i] + xw[i] / deg[node] + b[j];
  v += res1[(size_t)node * ldr1 + j];
  if (res2) v += res2[(size_t)node * ldr2 + j];
  out[(size_t)node * ldc + co + j] = v;
}

// ---------------- WMMA fp32 GEMM: C[16-row tile, 128 cols] ----------------
// A: [NN, K] row-major; W: [K, 128] row-major.
// Block = 256 threads = 8 waves; each wave owns one 16x16 column tile.
// A-frag layout (V_WMMA_F32_16X16X4_F32, ISA 7.12.2): lanes 0-15 hold K=0(v0),K=1(v1);
// lanes 16-31 hold K=2(v0),K=3(v1). B mirrors with N striped across lanes.
__global__ void __launch_bounds__(256)
gemm_wmma_kernel(const float* __restrict__ A, int K, const float* __restrict__ W,
                 const float* __restrict__ bias, float* __restrict__ C, int ldc, int co,
                 const float* __restrict__ res1, int ldr1,
                 const float* __restrict__ res2, int ldr2,
                 int acc, int relu) {
  __shared__ float As[16 * 256];  // 16 rows x K (K <= 256)
  int m0 = blockIdx.x * 16;

  for (int idx = threadIdx.x; idx < 16 * K; idx += 256) {
    int r = idx / K;
    int c = idx - r * K;
    As[idx] = A[(size_t)(m0 + r) * K + c];
  }
  __syncthreads();

  int lane = threadIdx.x & 31;
  int wave = threadIdx.x >> 5;  // 0..7 -> column tile
  int n0 = wave * 16;
  int row = lane & 15;          // M index for A, N index for B/C
  int hi = lane >> 4;           // 0 or 1

  v8f cacc = {};
  for (int k0 = 0; k0 < K; k0 += 4) {
    int ka = k0 + 2 * hi;
    v2f a, b;
    a.x = As[row * K + ka];
    a.y = As[row * K + ka + 1];
    b.x = W[(size_t)ka * HH + n0 + row];
    b.y = W[(size_t)(ka + 1) * HH + n0 + row];
    cacc = __builtin_amdgcn_wmma_f32_16x16x4_f32(false, a, false, b, (short)0, cacc,
                                                 false, false);
  }

  int col = n0 + row;
  float bv = bias ? bias[col] : 0.0f;
#pragma unroll
  for (int r = 0; r < 8; ++r) {
    int m = m0 + r + hi * 8;   // C/D layout: VGPR r -> M=r (lanes 0-15), M=r+8 (lanes 16-31)
    size_t off = (size_t)m * ldc + co + col;
    float v = cacc[r] + bv;
    if (acc) v += C[off];
    if (res1) v += res1[(size_t)m * ldr1 + col];
    if (res2) v += res2[(size_t)m * ldr2 + col];
    if (relu) v = fmaxf(v, 0.0f);
    C[off] = v;
  }
}

// ---------------- fused 4-head output: one wave32 per node ----------------
__global__ void head_kernel(const float* __restrict__ o1, const float* __restrict__ o2,
                            const float* __restrict__ o3, const float* __restrict__ o4,
                            const float* __restrict__ l1W, const float* __restrict__ l1b,
                            const float* __restrict__ l2W, const float* __restrict__ l2b,
                            const float* __restrict__ l3W, const float* __restrict__ l3b,
                            const float* __restrict__ l4W, const float* __restrict__ l4b,
                            const float* __restrict__ w1, const float* __restrict__ w2,
                            const float* __restrict__ w3, const float* __restrict__ w4,
                            float* __restrict__ res) {
  int wave = threadIdx.x >> 5;
  int lane = threadIdx.x & 31;
  int node = blockIdx.x * 8 + wave;
  if (node >= NN) return;
  float s1 = 0.f, s2 = 0.f, s3 = 0.f, s4 = 0.f;
  for (int k = lane; k < 128; k += 32) s1 += o1[(size_t)node * 128 + k] * l1W[k];
  for (int k = lane; k < 256; k += 32) {
    s2 += o2[(size_t)node * 256 + k] * l2W[k];
    s3 += o3[(size_t)node * 256 + k] * l3W[k];
    s4 += o4[(size_t)node * 256 + k] * l4W[k];
  }
  for (int off = 16; off > 0; off >>= 1) {
    s1 += __shfl_down(s1, off, 32);
    s2 += __shfl_down(s2, off, 32);
    s3 += __shfl_down(s3, off, 32);
    s4 += __shfl_down(s4, off, 32);
  }
  if (lane == 0)
    res[node] = (s1 + l1b[0]) * w1[0] + (s2 + l2b[0]) * w2[0] +
                (s3 + l3b[0]) * w3[0] + (s4 + l4b[0]) * w4[0];
}

// ---------------- host orchestration ----------------

static inline int cdiv(long long a, int b) { return (int)((a + b - 1) / b); }

static void run_gemm(hipStream_t s, const float* A, int K, const float* W, const float* bias,
                     float* C, int ldc, int co, const float* r1, int ldr1, const float* r2,
                     int ldr2, int acc, int relu) {
  gemm_wmma_kernel<<<NN / 16, 256, 0, s>>>(A, K, W, bias, C, ldc, co, r1, ldr1, r2, ldr2, acc,
                                           relu);
}

extern "C" void kernel_launch(void* const* d_in, const int* in_sizes, int n_in,
                              void* d_out, int out_size, void* d_ws, size_t ws_size,
                              hipStream_t stream) {
  (void)in_sizes; (void)n_in; (void)out_size; (void)ws_size;

  const float* x = (const float*)d_in[0];
  const int* ei = (const int*)d_in[1];
  const int* eia = (const int*)d_in[2];
  const int *src = ei, *dst = ei + EE;
  const int *srca = eia, *dsta = eia + EE;
  const float* fc1_W = (const float*)d_in[3];
  const float* fc1_b = (const float*)d_in[4];
  const float* sWl[3] = {(const float*)d_in[5], (const float*)d_in[8], (const float*)d_in[11]};
  const float* sbl[3] = {(const float*)d_in[6], (const float*)d_in[9], (const float*)d_in[12]};
  const float* sWr[3] = {(const float*)d_in[7], (const float*)d_in[10], (const float*)d_in[13]};
  const float* gW[3]  = {(const float*)d_in[14], (const float*)d_in[16], (const float*)d_in[18]};
  const float* gb[3]  = {(const float*)d_in[15], (const float*)d_in[17], (const float*)d_in[19]};
  const float* lW[4]  = {(const float*)d_in[20], (const float*)d_in[22], (const float*)d_in[24],
                         (const float*)d_in[26]};
  const float* lb[4]  = {(const float*)d_in[21], (const float*)d_in[23], (const float*)d_in[25],
                         (const float*)d_in[27]};
  const float* wv[4]  = {(const float*)d_in[28], (const float*)d_in[29], (const float*)d_in[30],
                         (const float*)d_in[31]};

  float* ws = (float*)d_ws;
  float* out1 = ws;                                  // [N,128]
  float* out2 = out1 + (size_t)NN * 128;             // [N,256]
  float* out3 = out2 + (size_t)NN * 256;             // [N,256]
  float* out4 = out3 + (size_t)NN * 256;             // [N,256]
  float* agg  = out4 + (size_t)NN * 256;             // [N,256] (also GCN scatter tmp)
  float* xw   = agg + (size_t)NN * 256;              // [N,128]
  float* degs = xw + (size_t)NN * 128;               // [N]
  float* degg = degs + NN;                           // [N]

  // ---- fc1: out1 = relu(x @ fc1_W + b) ----
  run_gemm(stream, x, 128, fc1_W, fc1_b, out1, 128, 0, nullptr, 0, nullptr, 0, 0, 1);

  // ---- three SAGE+GCN layers ----
  const float* X[3]   = {out1, out2, out3};
  const int    F[3]   = {128, 256, 256};
  float*       OUT[3] = {out2, out3, out4};

  for (int L = 0; L < 3; ++L) {
    const float* Xi = X[L];
    const int Fi = F[L];
    float* Oi = OUT[L];
    const float* x1a = (L == 0) ? nullptr : out2;        // x1 first half
    const float* x1b = (L == 0) ? nullptr : out2 + 128;  // x1 second half

    // --- SAGE half: OUT[:, :128] = mean_agg @ Wl + bl + Xi @ Wr + x0 (+ x1a) ---
    fill_kernel<<<cdiv(NN, 256), 256, 0, stream>>>(degs, NN, 0.f);
    deg_scatter_kernel<<<cdiv(EE, 256), 256, 0, stream>>>(dst, degs);
    fill_kernel<<<cdiv((long long)NN * Fi, 256), 256, 0, stream>>>(agg, NN * Fi, 0.f);
    sage_scatter_kernel<<<cdiv((long long)EE * (Fi / 4), 256), 256, 0, stream>>>(src, dst, Xi,
                                                                                 agg, Fi / 4);
    sage_div_kernel<<<cdiv((long long)NN * Fi, 256), 256, 0, stream>>>(agg, degs, Fi);
    run_gemm(stream, agg, Fi, sWl[L], sbl[L], Oi, 256, 0, nullptr, 0, nullptr, 0, 0, 0);
    run_gemm(stream, Xi, Fi, sWr[L], nullptr, Oi, 256, 0, out1, 128, x1a, 256, 1, 0);

    // --- GCN half: OUT[:, 128:] = scatter(norm*xw) + xw/deg + gb + x0 (+ x1b) ---
    fill_kernel<<<cdiv(NN, 256), 256, 0, stream>>>(degg, NN, 1.f);  // self-loop
    deg_scatter_kernel<<<cdiv(EE, 256), 256, 0, stream>>>(dsta, degg);
    run_gemm(stream, Xi, Fi, gW[L], nullptr, xw, 128, 0, nullptr, 0, nullptr, 0, 0, 0);
    fill_kernel<<<cdiv((long long)NN * 128, 256), 256, 0, stream>>>(agg, NN * 128, 0.f);
    gcn_scatter_kernel<<<cdiv((long long)EE * 32, 256), 256, 0, stream>>>(srca, dsta, xw, degg,
                                                                          agg);
    gcn_finish_kernel<<<cdiv((long long)NN * 128, 256), 256, 0, stream>>>(
        agg, xw, degg, gb[L], Oi, 256, 128, out1, 128, x1b, 256);
  }

  // ---- fused output heads -> d_out [N,1] ----
  head_kernel<<<cdiv(NN, 8), 256, 0, stream>>>(out1, out2, out3, out4,
                                               lW[0], lb[0], lW[1], lb[1],
                                               lW[2], lb[2], lW[3], lb[3],
                                               wv[0], wv[1], wv[2], wv[3],
                                               (float*)d_out);
}